// Coord_MultiStrip_Att_48326972015251
// MI455X (gfx1250) — compile-verified
//
#include <hip/hip_runtime.h>
#include <hip/hip_bf16.h>
#include <math.h>

// ---------------------------------------------------------------------------
// Shapes: B=32, C=256, H=W=64, MIP=8, O=256
// ws layout (floats): xh[512K] | xw[512K] | z[1M] | ah[512K] | aw[512K] = 12 MB
// ---------------------------------------------------------------------------

#define B_   32
#define C_   256
#define HW_  64
#define MIP_ 8
#define O_   256

typedef __attribute__((ext_vector_type(2))) float v2f;
typedef __attribute__((ext_vector_type(8))) float v8f;

// ---------------------------------------------------------------------------
// Kernel 1: per-(b,c) 64x64 tile -> row means (x_h) and column means (x_w).
// LDS tile padded to stride 65 so column reads are bank-conflict free.
// ---------------------------------------------------------------------------
__global__ void __launch_bounds__(256) reduce_hw_kernel(
    const float* __restrict__ x, float* __restrict__ xh, float* __restrict__ xw) {
  __shared__ float tile[64 * 65];
  const int bc  = blockIdx.x;          // 0 .. B*C-1
  const int tid = threadIdx.x;
  const int h   = tid >> 2;            // 0..63
  const int q   = tid & 3;             // 0..3 (16-column slab)
  const float* src = x + (size_t)bc * (HW_ * HW_);
#pragma unroll
  for (int i = 0; i < 4; ++i) {
    const int w = q * 16 + i * 4;
    const float4 v = *reinterpret_cast<const float4*>(src + h * HW_ + w);
    tile[h * 65 + w + 0] = v.x;
    tile[h * 65 + w + 1] = v.y;
    tile[h * 65 + w + 2] = v.z;
    tile[h * 65 + w + 3] = v.w;
  }
  __syncthreads();
  if (tid < 64) {                      // row mean -> x_h[b,c,h]
    float s = 0.f;
#pragma unroll
    for (int w = 0; w < 64; ++w) s += tile[tid * 65 + w];
    xh[(size_t)bc * HW_ + tid] = s * (1.0f / 64.0f);
  } else if (tid < 128) {              // column mean -> x_w[b,c,w]
    const int w = tid - 64;
    float s = 0.f;
#pragma unroll
    for (int hh = 0; hh < 64; ++hh) s += tile[hh * 65 + w];
    xw[(size_t)bc * HW_ + w] = s * (1.0f / 64.0f);
  }
}

// ---------------------------------------------------------------------------
// Kernel 2: strip pool: (x + dwconv3 + dwconv7)/3, then BN, for both strips.
// Output z[b, c, l] with l in [0,64) = H strip, [64,128) = W strip.
// One 128-thread block per (b,c).
// ---------------------------------------------------------------------------
__global__ void __launch_bounds__(128) strip_pool_kernel(
    const float* __restrict__ xh, const float* __restrict__ xw,
    const float* __restrict__ w3h, const float* __restrict__ w7h,
    const float* __restrict__ gh, const float* __restrict__ bh,
    const float* __restrict__ mh, const float* __restrict__ vh,
    const float* __restrict__ w3w, const float* __restrict__ w7w,
    const float* __restrict__ gw, const float* __restrict__ bw,
    const float* __restrict__ mw, const float* __restrict__ vw,
    float* __restrict__ z) {
  __shared__ float s[128];
  const int bc = blockIdx.x;
  const int c  = bc & (C_ - 1);
  const int l  = threadIdx.x;          // 0..127
  const int p  = l & 63;
  const bool isH = (l < 64);
  s[l] = (isH ? xh : xw)[(size_t)bc * HW_ + p];
  __syncthreads();

  const float* w3 = isH ? w3h : w3w;
  const float* w7 = isH ? w7h : w7w;
  const float g = isH ? gh[c] : gw[c];
  const float b = isH ? bh[c] : bw[c];
  const float m = isH ? mh[c] : mw[c];
  const float v = isH ? vh[c] : vw[c];
  const int base = isH ? 0 : 64;

  float acc = s[l];
#pragma unroll
  for (int j = -1; j <= 1; ++j) {      // k=3, zero pad
    const int pp = p + j;
    if (pp >= 0 && pp < 64) acc += w3[c * 3 + (j + 1)] * s[base + pp];
  }
#pragma unroll
  for (int j = -3; j <= 3; ++j) {      // k=7, zero pad
    const int pp = p + j;
    if (pp >= 0 && pp < 64) acc += w7[c * 7 + (j + 3)] * s[base + pp];
  }
  acc *= (1.0f / 3.0f);
  const float scale = g * __frsqrt_rn(v + 1e-5f);
  z[(size_t)bc * 128 + l] = (acc - m) * scale + b;
}

// ---------------------------------------------------------------------------
// Kernel 3 (WMMA): per-batch block of 256 threads (8 waves).
//  Phase 1: y = conv1_w(8x256) @ z_b(256x128) via V_WMMA_F32_16X16X4_F32
//           (M padded 8->16 with a 0/1 lane mask -> no EXEC divergence in the
//           hot loop), then bn1 + hswish, stored to LDS (8x128).
//  Phase 2: a_h = sigmoid(convh_w(256x8) @ y[:, :64]),
//           a_w = sigmoid(convw_w(256x8) @ y[:, 64:]) via the same WMMA.
// f32 A 16x4 layout: lanes 0-15 hold K={0,1}, lanes 16-31 hold K={2,3};
// f32 B 4x16 mirrors it with lanes holding N. C/D: VGPR r -> M=r | M=r+8.
// ---------------------------------------------------------------------------
__global__ void __launch_bounds__(256) attn_mm_kernel(
    const float* __restrict__ z,       // (B, 256, 128)
    const float* __restrict__ conv1_w, // (8, 256)
    const float* __restrict__ bn1_g, const float* __restrict__ bn1_b,
    const float* __restrict__ bn1_m, const float* __restrict__ bn1_v,
    const float* __restrict__ convh_w, // (256, 8)
    const float* __restrict__ convw_w, // (256, 8)
    float* __restrict__ ah, float* __restrict__ aw) {
  __shared__ float yLDS[MIP_ * 128];
  const int b    = blockIdx.x;
  const int tid  = threadIdx.x;
  const int wave = tid >> 5;
  const int lane = tid & 31;
  const int lm   = lane & 15;
  const int half = (lane < 16) ? 0 : 2;   // K sub-offset owned by this lane half

  const float* zb = z + (size_t)b * C_ * 128;

  // ---- Phase 1: wave w owns N-tile [w*16, w*16+16) of the 128 columns ----
  {
    const int n = wave * 16 + lm;
    // Branch-free zero padding of A rows 8..15: unconditional load from a
    // clamped row, scaled by a 0/1 mask (WMMA needs EXEC all-ones anyway).
    const float  amask = (lm < MIP_) ? 1.0f : 0.0f;
    const float* arow  = conv1_w + (size_t)(lm & (MIP_ - 1)) * C_;
    const float* bcol  = zb + n;
    v8f acc = {0.f, 0.f, 0.f, 0.f, 0.f, 0.f, 0.f, 0.f};
    for (int k0 = 0; k0 < C_; k0 += 4) {
      v2f a, bf;
      a.x  = amask * arow[k0 + half];
      a.y  = amask * arow[k0 + half + 1];
      bf.x = bcol[(size_t)(k0 + half) * 128];
      bf.y = bcol[(size_t)(k0 + half + 1) * 128];
      acc = __builtin_amdgcn_wmma_f32_16x16x4_f32(
          false, a, false, bf, (short)0, acc, false, false);
    }
    if (lane < 16) {                     // valid rows M=0..7 live in lanes 0-15
#pragma unroll
      for (int r = 0; r < MIP_; ++r) {
        const float sc = bn1_g[r] * __frsqrt_rn(bn1_v[r] + 1e-5f);
        float vv = (acc[r] - bn1_m[r]) * sc + bn1_b[r];
        float t = vv + 3.0f;
        t = fminf(fmaxf(t, 0.0f), 6.0f);
        vv = vv * t * (1.0f / 6.0f);     // hswish
        yLDS[r * 128 + n] = vv;
      }
    }
  }
  __syncthreads();

  // ---- Phase 2: 128 tile jobs = {h,w} x 16 M-tiles x 4 N-tiles ----
  for (int i = 0; i < 16; ++i) {
    const int j   = wave + 8 * i;        // bijective over 0..127
    const int sel = j >> 6;              // 0 = h branch, 1 = w branch
    const int rem = j & 63;
    const int mt  = rem >> 2;            // 0..15
    const int nt  = rem & 3;             // 0..3
    const float* cw = sel ? convw_w : convh_w;
    const int m = mt * 16 + lm;
    const int l = sel * 64 + nt * 16 + lm;
    v8f acc = {0.f, 0.f, 0.f, 0.f, 0.f, 0.f, 0.f, 0.f};
#pragma unroll
    for (int k0 = 0; k0 < MIP_; k0 += 4) {
      v2f a, bf;
      a.x  = cw[m * MIP_ + k0 + half];
      a.y  = cw[m * MIP_ + k0 + half + 1];
      bf.x = yLDS[(k0 + half) * 128 + l];
      bf.y = yLDS[(k0 + half + 1) * 128 + l];
      acc = __builtin_amdgcn_wmma_f32_16x16x4_f32(
          false, a, false, bf, (short)0, acc, false, false);
    }
    float* dst = sel ? aw : ah;
    const int n     = nt * 16 + lm;
    const int mbase = mt * 16 + ((lane < 16) ? 0 : 8);
#pragma unroll
    for (int r = 0; r < 8; ++r) {
      const float vv = 1.0f / (1.0f + __expf(-acc[r]));   // sigmoid
      dst[((size_t)b * O_ + (mbase + r)) * HW_ + n] = vv;
    }
  }
}

// ---------------------------------------------------------------------------
// Kernel 4: out = x * a_h[b,c,h] * a_w[b,c,w]  (float4, fully coalesced).
// idx enumerates B*C*H*(W/4) float4s: idx = bc*1024 + h*16 + w4.
// ---------------------------------------------------------------------------
__global__ void __launch_bounds__(256) apply_attn_kernel(
    const float* __restrict__ x, const float* __restrict__ ah,
    const float* __restrict__ aw, float* __restrict__ out) {
  const size_t idx = (size_t)blockIdx.x * blockDim.x + threadIdx.x;
  const int w4      = (int)(idx & 15);
  const size_t rest = idx >> 4;
  const int h       = (int)(rest & 63);
  const size_t bc   = rest >> 6;

  const float  sa = ah[bc * HW_ + h];
  const float4 wa = reinterpret_cast<const float4*>(aw + bc * HW_)[w4];
  const float4 xv = reinterpret_cast<const float4*>(x)[idx];
  float4 o;
  o.x = xv.x * sa * wa.x;
  o.y = xv.y * sa * wa.y;
  o.z = xv.z * sa * wa.z;
  o.w = xv.w * sa * wa.w;
  reinterpret_cast<float4*>(out)[idx] = o;
}

// ---------------------------------------------------------------------------
extern "C" void kernel_launch(void* const* d_in, const int* in_sizes, int n_in,
                              void* d_out, int out_size, void* d_ws, size_t ws_size,
                              hipStream_t stream) {
  const float* x        = (const float*)d_in[0];
  const float* sph_w3   = (const float*)d_in[1];
  const float* sph_w7   = (const float*)d_in[2];
  const float* sph_g    = (const float*)d_in[3];
  const float* sph_b    = (const float*)d_in[4];
  const float* sph_m    = (const float*)d_in[5];
  const float* sph_v    = (const float*)d_in[6];
  const float* spw_w3   = (const float*)d_in[7];
  const float* spw_w7   = (const float*)d_in[8];
  const float* spw_g    = (const float*)d_in[9];
  const float* spw_b    = (const float*)d_in[10];
  const float* spw_m    = (const float*)d_in[11];
  const float* spw_v    = (const float*)d_in[12];
  const float* conv1_w  = (const float*)d_in[13];
  const float* bn1_g    = (const float*)d_in[14];
  const float* bn1_b    = (const float*)d_in[15];
  const float* bn1_m    = (const float*)d_in[16];
  const float* bn1_v    = (const float*)d_in[17];
  const float* convh_w  = (const float*)d_in[18];
  const float* convw_w  = (const float*)d_in[19];
  float* out = (float*)d_out;

  float* ws = (float*)d_ws;
  float* xh = ws;                        // B*C*64  = 524288 floats
  float* xw = xh + (size_t)B_ * C_ * HW_;
  float* z  = xw + (size_t)B_ * C_ * HW_;      // B*C*128 = 1048576 floats
  float* ah = z  + (size_t)B_ * C_ * 128;      // B*O*64
  float* aw = ah + (size_t)B_ * O_ * HW_;

  // 1) H/W mean reductions of x (reads 128 MB once)
  reduce_hw_kernel<<<B_ * C_, 256, 0, stream>>>(x, xh, xw);

  // 2) strip pooling + BN -> z (B, 256, 128)
  strip_pool_kernel<<<B_ * C_, 128, 0, stream>>>(
      xh, xw,
      sph_w3, sph_w7, sph_g, sph_b, sph_m, sph_v,
      spw_w3, spw_w7, spw_g, spw_b, spw_m, spw_v, z);

  // 3) WMMA matmuls + bn1 + hswish + sigmoid -> a_h, a_w
  attn_mm_kernel<<<B_, 256, 0, stream>>>(
      z, conv1_w, bn1_g, bn1_b, bn1_m, bn1_v, convh_w, convw_w, ah, aw);

  // 4) out = x * a_h * a_w (reads 128 MB + writes 128 MB)
  const size_t n4 = (size_t)B_ * C_ * HW_ * HW_ / 4;   // 8,388,608 float4s
  apply_attn_kernel<<<(int)(n4 / 256), 256, 0, stream>>>(x, ah, aw, out);
}